// Memory_14903536517587
// MI455X (gfx1250) — compile-verified
//
#include <hip/hip_runtime.h>
#include <math.h>

// ---------------------------------------------------------------------------
// DNC memory step for MI455X (gfx1250, wave32).
// N=8192, W=128, R=4.  Traffic floor = link read + new_link write = 512MB
// -> ~22us @ 23.3TB/s.  Single fused pass over link; fp32 WMMA (16x16x4)
// for the cosine-similarity scores to keep exact fp32 numerics.
// ---------------------------------------------------------------------------

typedef float vf2 __attribute__((ext_vector_type(2)));
typedef float vf4 __attribute__((ext_vector_type(4)));
typedef float vf8 __attribute__((ext_vector_type(8)));

constexpr int N = 8192;
constexpr int W = 128;
constexpr int R = 4;
constexpr float EPS = 1e-8f;

// link-pass geometry
constexpr int LB    = 64;       // blocks over rows
constexpr int LT    = 1024;     // threads per block (32 waves)
constexpr int LROWS = N / LB;   // 128 rows per block

// ---- workspace layout (floats) --------------------------------------------
constexpr size_t OFF_KN    = 0;                       // 8  : knorm[0..3], wknorm@4
constexpr size_t OFF_SC    = 8;                       // 5N : raw scores (r=0..3 read, r=4 write)
constexpr size_t OFF_PROB  = OFF_SC    + 5 * N;       // 5N : softmax probs
constexpr size_t OFF_SORTU = OFF_PROB  + 5 * N;       // N  : sorted usage
constexpr size_t OFF_ORDER = OFF_SORTU + N;           // N  : order indices (int)
constexpr size_t OFF_ALLOC = OFF_ORDER + N;           // N  : allocation weighting
constexpr size_t OFF_WW    = OFF_ALLOC + N;           // N  : write weighting
constexpr size_t OFF_SUMWW = OFF_WW    + N;           // 4  : sum(write_w)
constexpr size_t OFF_BW    = OFF_SUMWW + 4;           // 4N : backward_w
constexpr size_t OFF_RW    = OFF_BW    + 4 * N;       // 4N : read_w
constexpr size_t OFF_RVP   = OFF_RW    + 4 * N;       // 64*512 : read-vector partials
constexpr size_t OFF_FPART = OFF_RVP   + 64 * 512;    // LB*4*N : forward partials
constexpr size_t WS_FLOATS = OFF_FPART + (size_t)LB * 4 * N;   // ~9.3 MB

// ---- d_out layout (floats, concatenated return order) ----------------------
constexpr size_t OUT_RV   = 0;                        // R*W
constexpr size_t OUT_MEM  = OUT_RV  + (size_t)R * W;  // N*W
constexpr size_t OUT_USE  = OUT_MEM + (size_t)N * W;  // N
constexpr size_t OUT_PREC = OUT_USE + N;              // N
constexpr size_t OUT_LINK = OUT_PREC + N;             // N*N

// ===========================================================================
// K0: key norms (4 read keys + write key) -> ws[OFF_KN..OFF_KN+4]
// ===========================================================================
__global__ void k_key_norms(const float* __restrict__ rk,
                            const float* __restrict__ wk,
                            float* __restrict__ kn) {
    int wave = threadIdx.x >> 5, lane = threadIdx.x & 31;
    if (wave >= 5) return;
    const float* p = (wave < 4) ? (rk + wave * W) : wk;
    float s = 0.f;
    for (int j = lane; j < W; j += 32) s += p[j] * p[j];
    for (int off = 16; off > 0; off >>= 1) s += __shfl_xor(s, off, 32);
    if (lane == 0) kn[wave] = sqrtf(s);
}

// ===========================================================================
// K1: WMMA fp32 similarity scores.  One wave per 16 memory rows.
// A (16x128, LDS): rows 0..3 read_keys, row 4 write_key, rows 5..15 zero.
// B (128x16) = memory-row tile (transposed access); per-lane float2 loads
// also feed the row-norm (sum-of-squares, finished via shfl_xor(16)).
// D = A x B accumulated with 32 chained v_wmma_f32_16x16x4_f32.
// ===========================================================================
__global__ void k_scores_wmma(const float* __restrict__ memory,
                              const float* __restrict__ read_keys,
                              const float* __restrict__ write_key,
                              const float* __restrict__ read_strengths,
                              const float* __restrict__ write_strength,
                              const float* __restrict__ knorms,
                              float* __restrict__ scores) {
    __shared__ float A[16 * 128];
    int tid = threadIdx.x;
    for (int idx = tid; idx < 16 * 128; idx += 256) {
        int m = idx >> 7, k = idx & 127;
        float v = 0.f;
        if (m < 4)      v = read_keys[m * W + k];
        else if (m == 4) v = write_key[k];
        A[idx] = v;
    }
    __syncthreads();

    int wave = tid >> 5, lane = tid & 31;
    int half = lane >> 4, mr = lane & 15;
    size_t n0 = ((size_t)blockIdx.x * 8 + wave) * 16;

    const float* mrow = memory + (n0 + (size_t)mr) * W + 2 * half;
    const float* arow = A + mr * 128 + 2 * half;

    vf8 acc = {0.f, 0.f, 0.f, 0.f, 0.f, 0.f, 0.f, 0.f};
    float ssq = 0.f;
#pragma unroll 8
    for (int s = 0; s < 32; ++s) {
        // A 16x4 layout: lanes0-15 hold K={0,1}, lanes16-31 K={2,3}
        vf2 a = *(const vf2*)(arow + 4 * s);
        // B 4x16 layout mirrors it; per-lane float2 from memory row n0+mr
        vf2 b = *(const vf2*)(mrow + 4 * s);
        ssq += b.x * b.x + b.y * b.y;
        acc = __builtin_amdgcn_wmma_f32_16x16x4_f32(false, a, false, b,
                                                    (short)0, acc, false, false);
    }
    ssq += __shfl_xor(ssq, 16, 32);       // lanes l and l^16 cover one row
    float mn = sqrtf(ssq);

    if (lane < 16) {                       // D vgpr i, lanes 0-15 -> M=i
        size_t n = n0 + (size_t)mr;
#pragma unroll
        for (int r = 0; r < 4; ++r)
            scores[(size_t)r * N + n] =
                acc[r] * read_strengths[r] / fmaxf(knorms[r] * mn, EPS);
        scores[(size_t)4 * N + n] =
            acc[4] * write_strength[0] / fmaxf(mn * knorms[4], EPS);
    }
}

// ===========================================================================
// K2: softmax over N for 5 score rows (blockIdx.x = row)
// ===========================================================================
__global__ void k_softmax5(const float* __restrict__ scores,
                           float* __restrict__ probs) {
    __shared__ float red[1024];
    const float* s = scores + (size_t)blockIdx.x * N;
    float* o = probs + (size_t)blockIdx.x * N;
    int tid = threadIdx.x;

    float m = -3.4e38f;
    for (int j = tid; j < N; j += 1024) m = fmaxf(m, s[j]);
    red[tid] = m; __syncthreads();
    for (int st = 512; st > 0; st >>= 1) {
        if (tid < st) red[tid] = fmaxf(red[tid], red[tid + st]);
        __syncthreads();
    }
    float mx = red[0]; __syncthreads();

    float sum = 0.f;
    for (int j = tid; j < N; j += 1024) sum += expf(s[j] - mx);
    red[tid] = sum; __syncthreads();
    for (int st = 512; st > 0; st >>= 1) {
        if (tid < st) red[tid] += red[tid + st];
        __syncthreads();
    }
    float inv = 1.f / red[0]; __syncthreads();

    for (int j = tid; j < N; j += 1024) o[j] = expf(s[j] - mx) * inv;
}

// ===========================================================================
// K3a: single-block bitonic sort of usage (stable via packed index)
// ===========================================================================
__global__ void k_sort_usage(const float* __restrict__ usage,
                             float* __restrict__ sortedU,
                             int* __restrict__ order) {
    __shared__ unsigned long long key[8192];   // 64KB LDS
    int tid = threadIdx.x;
    for (int j = tid; j < N; j += 1024) {
        unsigned b = __float_as_uint(usage[j]);
        b = (b & 0x80000000u) ? ~b : (b | 0x80000000u);   // total order
        key[j] = ((unsigned long long)b << 32) | (unsigned)j;
    }
    __syncthreads();
    for (int k = 2; k <= N; k <<= 1)
        for (int j = k >> 1; j > 0; j >>= 1) {
            for (int i = tid; i < N; i += 1024) {
                int l = i ^ j;
                if (l > i) {
                    bool asc = ((i & k) == 0);
                    unsigned long long a = key[i], b2 = key[l];
                    if ((a > b2) == asc) { key[i] = b2; key[l] = a; }
                }
            }
            __syncthreads();
        }
    for (int j = tid; j < N; j += 1024) {
        unsigned long long kk = key[j];
        unsigned b = (unsigned)(kk >> 32);
        b = (b & 0x80000000u) ? (b & 0x7fffffffu) : ~b;
        sortedU[j] = __uint_as_float(b);
        order[j] = (int)(kk & 0xffffffffu);
    }
}

// ===========================================================================
// K3b: exclusive multiply-scan + gather allocation_w[i]=alloc_sorted[order[i]]
// ===========================================================================
__global__ void k_alloc_scan(const float* __restrict__ sortedU,
                             const int* __restrict__ order,
                             float* __restrict__ allocW) {
    __shared__ float cp[8192];                 // 32KB LDS
    int tid = threadIdx.x;
    for (int j = tid; j < N; j += 1024) cp[j] = sortedU[j];
    __syncthreads();
    for (int off = 1; off < N; off <<= 1) {    // inclusive cumprod
        float t[8]; int c = 0;
        for (int j = tid; j < N; j += 1024) t[c++] = (j >= off) ? cp[j - off] : 1.f;
        __syncthreads();
        c = 0;
        for (int j = tid; j < N; j += 1024) cp[j] *= t[c++];
        __syncthreads();
    }
    float a[8]; int c = 0;
    for (int j = tid; j < N; j += 1024) {
        float ex = (j == 0) ? 1.f : cp[j - 1];
        a[c++] = (1.f - sortedU[j]) * ex;
    }
    __syncthreads();
    c = 0;
    for (int j = tid; j < N; j += 1024) cp[j] = a[c++];   // alloc_sorted in LDS
    __syncthreads();
    for (int j = tid; j < N; j += 1024) allocW[j] = cp[order[j]];  // gather (ref!)
}

// ===========================================================================
// K4: write weighting + sum(write_w)  (single block -> deterministic)
// ===========================================================================
__global__ void k_write_w(const float* __restrict__ allocW,
                          const float* __restrict__ probs,
                          const float* __restrict__ agate,
                          const float* __restrict__ wgate,
                          float* __restrict__ ww,
                          float* __restrict__ sumOut) {
    __shared__ float red[1024];
    int tid = threadIdx.x;
    float a = agate[0], g = wgate[0];
    const float* wcw = probs + (size_t)4 * N;
    float loc = 0.f;
    for (int j = tid; j < N; j += 1024) {
        float w = g * (a * allocW[j] + (1.f - a) * wcw[j]);
        ww[j] = w; loc += w;
    }
    red[tid] = loc; __syncthreads();
    for (int st = 512; st > 0; st >>= 1) {
        if (tid < st) red[tid] += red[tid + st];
        __syncthreads();
    }
    if (tid == 0) sumOut[0] = red[0];
}

// ===========================================================================
// K5: fused link pass.  64 blocks x 1024 threads, 128 rows/block.
//   - backward_w[r,n]   = sum_m link[n,m]*prev[r,m]  (wave shfl + LDS tail)
//   - forward partials  = sum_{n in tile} prev[r,n]*link[n,m]  (regs -> ws)
//   - new_link[n,m]     = (1-ww[n]-ww[m])*link + ww[n]*prec[m] (non-temporal)
// Deterministic: no float atomics anywhere.
// ===========================================================================
__global__ void __launch_bounds__(LT)
k_link_fused(const float* __restrict__ link,
             const float* __restrict__ prev,
             const float* __restrict__ precedence,
             const float* __restrict__ ww,
             float* __restrict__ newlink,
             float* __restrict__ bwOut,
             float* __restrict__ fpart) {
    int tid = threadIdx.x, wave = tid >> 5, lane = tid & 31;
    size_t rowBase = (size_t)blockIdx.x * LROWS;
    int c[2] = { tid * 4, tid * 4 + 4096 };

    vf4 prevc[2][4], wwc[2], prc[2], facc[2][4];
#pragma unroll
    for (int g = 0; g < 2; ++g) {
#pragma unroll
        for (int r = 0; r < 4; ++r) {
            prevc[g][r] = *(const vf4*)(prev + (size_t)r * N + c[g]);
            facc[g][r] = (vf4){0.f, 0.f, 0.f, 0.f};
        }
        wwc[g] = *(const vf4*)(ww + c[g]);
        prc[g] = *(const vf4*)(precedence + c[g]);
    }

    __shared__ vf4 red[8][32];

    for (int i = 0; i < LROWS; ++i) {
        size_t n = rowBase + i;
        float wwn = ww[n];
        float p0 = prev[0 * N + n], p1 = prev[1 * N + n];
        float p2 = prev[2 * N + n], p3 = prev[3 * N + n];
        vf4 bw = {0.f, 0.f, 0.f, 0.f};
#pragma unroll
        for (int g = 0; g < 2; ++g) {
            const vf4* lp = (const vf4*)(link + n * N + c[g]);
            vf4 L = __builtin_nontemporal_load(lp);
            facc[g][0] += p0 * L;  facc[g][1] += p1 * L;
            facc[g][2] += p2 * L;  facc[g][3] += p3 * L;
            bw.x += L.x*prevc[g][0].x + L.y*prevc[g][0].y + L.z*prevc[g][0].z + L.w*prevc[g][0].w;
            bw.y += L.x*prevc[g][1].x + L.y*prevc[g][1].y + L.z*prevc[g][1].z + L.w*prevc[g][1].w;
            bw.z += L.x*prevc[g][2].x + L.y*prevc[g][2].y + L.z*prevc[g][2].z + L.w*prevc[g][2].w;
            bw.w += L.x*prevc[g][3].x + L.y*prevc[g][3].y + L.z*prevc[g][3].z + L.w*prevc[g][3].w;
            vf4 NL = (1.f - wwn - wwc[g]) * L + wwn * prc[g];
            __builtin_nontemporal_store(NL, (vf4*)(newlink + n * N + c[g]));
        }
        // wave32 reduction of the backward partial
        for (int off = 16; off > 0; off >>= 1) {
            bw.x += __shfl_xor(bw.x, off, 32);
            bw.y += __shfl_xor(bw.y, off, 32);
            bw.z += __shfl_xor(bw.z, off, 32);
            bw.w += __shfl_xor(bw.w, off, 32);
        }
        if (lane == 0) red[i & 7][wave] = bw;
        if ((i & 7) == 7) {                    // batched tail: 2 barriers / 8 rows
            __syncthreads();
            if (tid < 8) {
                vf4 s = {0.f, 0.f, 0.f, 0.f};
#pragma unroll
                for (int w2 = 0; w2 < 32; ++w2) s += red[tid][w2];
                size_t nr = rowBase + (size_t)(i & ~7) + tid;
                bwOut[0 * N + nr] = s.x;  bwOut[1 * N + nr] = s.y;
                bwOut[2 * N + nr] = s.z;  bwOut[3 * N + nr] = s.w;
            }
            __syncthreads();
        }
    }
#pragma unroll
    for (int g = 0; g < 2; ++g)
#pragma unroll
        for (int r = 0; r < 4; ++r)
            *(vf4*)(fpart + ((size_t)(blockIdx.x * 4 + r)) * N + c[g]) = facc[g][r];
}

// ===========================================================================
// K6: read_w, psi, new_usage, new_precedence  (reduces forward partials)
// ===========================================================================
__global__ void k_finalize_aux(const float* __restrict__ fpart,
                               const float* __restrict__ bwW,
                               const float* __restrict__ probs,
                               const float* __restrict__ modes,
                               const float* __restrict__ freeg,
                               const float* __restrict__ usage,
                               const float* __restrict__ precedence,
                               const float* __restrict__ wwW,
                               const float* __restrict__ sumWW,
                               float* __restrict__ rwOut,
                               float* __restrict__ outUsage,
                               float* __restrict__ outPrec) {
    int n = blockIdx.x * 256 + threadIdx.x;
    float fw[4] = {0.f, 0.f, 0.f, 0.f};
    for (int b = 0; b < LB; ++b)
#pragma unroll
        for (int r = 0; r < 4; ++r)
            fw[r] += fpart[((size_t)(b * 4 + r)) * N + n];

    float psi = 1.f;
#pragma unroll
    for (int r = 0; r < 4; ++r) {
        float bw = bwW[(size_t)r * N + n];
        float cw = probs[(size_t)r * N + n];
        float rw = modes[r * 3 + 0] * bw + modes[r * 3 + 1] * cw + modes[r * 3 + 2] * fw[r];
        rwOut[(size_t)r * N + n] = rw;
        psi *= (1.f - rw * freeg[r]);
    }
    float u = usage[n], w = wwW[n];
    outUsage[n] = (u + w - u * w) * psi;
    outPrec[n]  = (1.f - sumWW[0]) * precedence[n] + w;
}

// ===========================================================================
// K7: new_memory + read-vector partials (64 blocks x 128 threads)
// ===========================================================================
__global__ void k_memory_update(const float* __restrict__ memory,
                                const float* __restrict__ wwW,
                                const float* __restrict__ rwW,
                                const float* __restrict__ erase,
                                const float* __restrict__ wvec,
                                float* __restrict__ outMem,
                                float* __restrict__ rvPart) {
    int w = threadIdx.x;
    float e = erase[w], v = wvec[w];
    float rv[4] = {0.f, 0.f, 0.f, 0.f};
    size_t nb = (size_t)blockIdx.x * 128;
    for (int i = 0; i < 128; ++i) {
        size_t n = nb + i;
        float m = memory[n * W + w];
        float wn = wwW[n];
        __builtin_nontemporal_store(m * (1.f - wn * e) + wn * v, outMem + n * W + w);
#pragma unroll
        for (int r = 0; r < 4; ++r) rv[r] += rwW[(size_t)r * N + n] * m;
    }
#pragma unroll
    for (int r = 0; r < 4; ++r)
        rvPart[(size_t)blockIdx.x * 512 + r * W + w] = rv[r];
}

// K8: reduce read-vector partials (deterministic fixed-order sum)
__global__ void k_rv_reduce(const float* __restrict__ rvPart,
                            float* __restrict__ outRV) {
    int t = threadIdx.x;       // 512 = R*W
    float s = 0.f;
    for (int b = 0; b < LB; ++b) s += rvPart[(size_t)b * 512 + t];
    outRV[t] = s;
}

// ===========================================================================
extern "C" void kernel_launch(void* const* d_in, const int* in_sizes, int n_in,
                              void* d_out, int out_size, void* d_ws, size_t ws_size,
                              hipStream_t stream) {
    const float* memory      = (const float*)d_in[0];
    const float* usage       = (const float*)d_in[1];
    const float* precedence  = (const float*)d_in[2];
    const float* link        = (const float*)d_in[3];
    const float* prev_rw     = (const float*)d_in[4];
    const float* read_keys   = (const float*)d_in[5];
    const float* read_str    = (const float*)d_in[6];
    const float* read_modes  = (const float*)d_in[7];
    const float* write_key   = (const float*)d_in[8];
    const float* write_str   = (const float*)d_in[9];
    const float* alloc_gate  = (const float*)d_in[10];
    const float* write_gate  = (const float*)d_in[11];
    const float* erase_vec   = (const float*)d_in[12];
    const float* write_vec   = (const float*)d_in[13];
    const float* free_gate   = (const float*)d_in[14];

    float* out = (float*)d_out;
    float* ws  = (float*)d_ws;

    float* wsKN    = ws + OFF_KN;
    float* wsSC    = ws + OFF_SC;
    float* wsPROB  = ws + OFF_PROB;
    float* wsSORTU = ws + OFF_SORTU;
    int*   wsORDER = (int*)(ws + OFF_ORDER);
    float* wsALLOC = ws + OFF_ALLOC;
    float* wsWW    = ws + OFF_WW;
    float* wsSUM   = ws + OFF_SUMWW;
    float* wsBW    = ws + OFF_BW;
    float* wsRW    = ws + OFF_RW;
    float* wsRVP   = ws + OFF_RVP;
    float* wsFP    = ws + OFF_FPART;

    k_key_norms   <<<1, 160, 0, stream>>>(read_keys, write_key, wsKN);
    k_scores_wmma <<<N / 128, 256, 0, stream>>>(memory, read_keys, write_key,
                                                read_str, write_str, wsKN, wsSC);
    k_softmax5    <<<5, 1024, 0, stream>>>(wsSC, wsPROB);
    k_sort_usage  <<<1, 1024, 0, stream>>>(usage, wsSORTU, wsORDER);
    k_alloc_scan  <<<1, 1024, 0, stream>>>(wsSORTU, wsORDER, wsALLOC);
    k_write_w     <<<1, 1024, 0, stream>>>(wsALLOC, wsPROB, alloc_gate, write_gate,
                                           wsWW, wsSUM);
    k_link_fused  <<<LB, LT, 0, stream>>>(link, prev_rw, precedence, wsWW,
                                          out + OUT_LINK, wsBW, wsFP);
    k_finalize_aux<<<N / 256, 256, 0, stream>>>(wsFP, wsBW, wsPROB, read_modes,
                                                free_gate, usage, precedence,
                                                wsWW, wsSUM, wsRW,
                                                out + OUT_USE, out + OUT_PREC);
    k_memory_update<<<LB, W, 0, stream>>>(memory, wsWW, wsRW, erase_vec, write_vec,
                                          out + OUT_MEM, wsRVP);
    k_rv_reduce   <<<1, R * W, 0, stream>>>(wsRVP, out + OUT_RV);
}